// TAB_Attention_AF_67903432949953
// MI455X (gfx1250) — compile-verified
//
#include <hip/hip_runtime.h>
#include <math.h>

// Problem dims (from reference)
#define NB   32      // batch
#define SL   2048    // seq len
#define DMODEL 128   // d_model
#define WIN  64      // local window (|i-j| < WIN)
#define BAND_K 144   // padded band width per 16-row tile: [-64, +79], multiple of 4
#define NTILE (SL/16)   // 128 row-tiles per batch

typedef __attribute__((ext_vector_type(2))) float v2f;
typedef __attribute__((ext_vector_type(8))) float v8f;

// D = A(16x4) * B(4x16) + C  (fp32 WMMA, CDNA5)
__device__ __forceinline__ v8f wmma4(v2f a, v2f b, v8f c) {
  return __builtin_amdgcn_wmma_f32_16x16x4_f32(
      /*neg_a=*/false, a, /*neg_b=*/false, b,
      /*c_mod=*/(short)0, c, /*reuse_a=*/false, /*reuse_b=*/false);
}

// ---------------------------------------------------------------------------
// Kernel 1: band weights  b[i,j] = exp(pos_bias[i,j]) - 1 inside band, else 0.
// Stored compactly per 16-row tile: Wb[tile][m][kidx], j = 16*tile - WIN + kidx.
// Amortizes the exp() over all 32 batches.
// ---------------------------------------------------------------------------
__global__ void k_band_w(const float* __restrict__ pb, float* __restrict__ Wb) {
  int tile = blockIdx.x;
  for (int e = threadIdx.x; e < 16 * BAND_K; e += blockDim.x) {
    int m    = e / BAND_K;
    int kidx = e - m * BAND_K;
    int i = tile * 16 + m;
    int j = tile * 16 - WIN + kidx;
    float v = 0.0f;
    if (j >= 0 && j < SL) {
      int dlt = i - j; if (dlt < 0) dlt = -dlt;
      if (dlt < WIN) v = __expf(pb[(size_t)i * SL + j]) - 1.0f;
    }
    Wb[(size_t)tile * (16 * BAND_K) + e] = v;
  }
}

// ---------------------------------------------------------------------------
// Kernel 2: QKV projection. 3 waves per block: wave0->Q, wave1->K, wave2->V.
// Each wave: one 16-row tile x all 8 column tiles, K-loop of 32 x wmma 16x16x4.
// Outputs: sq = sigmoid(q), ek = exp(k), Xv = ek * v, plus per-tile column
// partial sums PK/PX (deterministic; reduced by k_colsum).
// ---------------------------------------------------------------------------
__global__ __launch_bounds__(96) void k_proj_qkv(
    const float* __restrict__ x,
    const float* __restrict__ Wq, const float* __restrict__ bq,
    const float* __restrict__ Wk, const float* __restrict__ bk,
    const float* __restrict__ Wv, const float* __restrict__ bv,
    float* __restrict__ sq, float* __restrict__ ek, float* __restrict__ Xv,
    float* __restrict__ PK, float* __restrict__ PX) {
  __shared__ float lds_ek[16 * DMODEL];

  const int wid  = threadIdx.x >> 5;       // 0:Q 1:K 2:V
  const int lane = threadIdx.x & 31;
  const int m    = lane & 15;              // A row / B,C column
  const int kh   = lane >> 4;              // K-half select

  const int rt = blockIdx.x;               // global 16-row tile id
  const int n  = rt >> 7;                  // NTILE = 128 tiles per batch
  const int l0 = (rt & (NTILE - 1)) << 4;

  const float* Wm   = (wid == 0) ? Wq : (wid == 1) ? Wk : Wv;
  const float* bias = (wid == 0) ? bq : (wid == 1) ? bk : bv;
  const float* xb   = x + (size_t)n * DMODEL * SL;

  v8f acc[8] = {};
  for (int k0 = 0; k0 < DMODEL; k0 += 4) {
    const int c = k0 + 2 * kh;
    // A fragment: xt[l0+m, c..c+1] = x[n, c..c+1, l0+m]  (stride-L apart)
    float a0 = xb[(size_t)c * SL + l0 + m];
    float a1 = xb[(size_t)(c + 1) * SL + l0 + m];
    v2f a = {a0, a1};
    const float* w0 = Wm + (size_t)c * DMODEL + m;
    const float* w1 = w0 + DMODEL;
#pragma unroll
    for (int jt = 0; jt < 8; jt++) {
      v2f b = {w0[jt * 16], w1[jt * 16]};
      acc[jt] = wmma4(a, b, acc[jt]);
    }
  }

  const size_t rowbase = ((size_t)n * SL + l0) * DMODEL;

  if (wid == 1) {                          // K wave: ek = exp(k), before barrier
#pragma unroll
    for (int jt = 0; jt < 8; jt++) {
      const int d = jt * 16 + m;
      float cs = 0.0f;
#pragma unroll
      for (int r = 0; r < 8; r++) {
        const int row = r + 8 * kh;
        float e = __expf(acc[jt][r] + bias[d]);
        ek[rowbase + (size_t)row * DMODEL + d] = e;
        lds_ek[row * DMODEL + d] = e;
        cs += e;
      }
      float tot = cs + __shfl_xor(cs, 16, 32);   // combine row-halves
      if (kh == 0) PK[(size_t)rt * DMODEL + d] = tot;
    }
  }
  __syncthreads();

  if (wid == 0) {                          // Q wave: sigmoid(q)
#pragma unroll
    for (int jt = 0; jt < 8; jt++) {
      const int d = jt * 16 + m;
#pragma unroll
      for (int r = 0; r < 8; r++) {
        const int row = r + 8 * kh;
        float v = acc[jt][r] + bias[d];
        sq[rowbase + (size_t)row * DMODEL + d] = 1.0f / (1.0f + __expf(-v));
      }
    }
  } else if (wid == 2) {                   // V wave: X = ek * v (reads LDS)
#pragma unroll
    for (int jt = 0; jt < 8; jt++) {
      const int d = jt * 16 + m;
      float cs = 0.0f;
#pragma unroll
      for (int r = 0; r < 8; r++) {
        const int row = r + 8 * kh;
        float e  = lds_ek[row * DMODEL + d];
        float xv = e * (acc[jt][r] + bias[d]);
        Xv[rowbase + (size_t)row * DMODEL + d] = xv;
        cs += xv;
      }
      float tot = cs + __shfl_xor(cs, 16, 32);
      if (kh == 0) PX[(size_t)rt * DMODEL + d] = tot;
    }
  }
}

// ---------------------------------------------------------------------------
// Kernel 3: reduce per-tile column partials -> S_K[n,d], S_X[n,d] (deterministic)
// ---------------------------------------------------------------------------
__global__ void k_colsum(const float* __restrict__ PK, const float* __restrict__ PX,
                         float* __restrict__ SK, float* __restrict__ SX) {
  int idx = blockIdx.x * blockDim.x + threadIdx.x;   // over NB*DMODEL
  if (idx >= NB * DMODEL) return;
  int n = idx / DMODEL, d = idx - n * DMODEL;
  const float* pk = PK + ((size_t)n * NTILE) * DMODEL + d;
  const float* px = PX + ((size_t)n * NTILE) * DMODEL + d;
  float sk = 0.0f, sx = 0.0f;
  for (int t = 0; t < NTILE; t++) {
    sk += pk[(size_t)t * DMODEL];
    sx += px[(size_t)t * DMODEL];
  }
  SK[idx] = sk;
  SX[idx] = sx;
}

// ---------------------------------------------------------------------------
// Kernel 4: banded aggregation. Per block: one (n, 16-row tile); 2 waves, each
// owning 4 of the 8 d-tiles for BOTH num and den (shared A = band weights).
// y = sigmoid(q) * (S_X + band*X) / (S_K + band*ek)
// ---------------------------------------------------------------------------
__global__ __launch_bounds__(64) void k_band_attn(
    const float* __restrict__ Wb, const float* __restrict__ Xv,
    const float* __restrict__ ek, const float* __restrict__ sq,
    const float* __restrict__ SK, const float* __restrict__ SX,
    float* __restrict__ y) {
  const int wid  = threadIdx.x >> 5;
  const int lane = threadIdx.x & 31;
  const int m    = lane & 15;
  const int kh   = lane >> 4;

  const int bt   = blockIdx.x;
  const int n    = bt >> 7;
  const int tile = bt & (NTILE - 1);
  const int I    = tile << 4;
  const int j0   = I - WIN;

  const float* wbase = Wb + (size_t)tile * (16 * BAND_K) + (size_t)m * BAND_K;
  const float* Xn = Xv + (size_t)n * SL * DMODEL;
  const float* Kn = ek + (size_t)n * SL * DMODEL;

  v8f nacc[4] = {}, dacc[4] = {};
  for (int kk = 0; kk < BAND_K; kk += 4) {
    // A fragment: band weights, two adjacent kidx -> one b64 load
    v2f a = *(const v2f*)(wbase + kk + 2 * kh);
    const int j = j0 + kk + 2 * kh;
    const bool in0 = ((unsigned)j < (unsigned)SL);
    const bool in1 = ((unsigned)(j + 1) < (unsigned)SL);
    const size_t r0 = (size_t)j * DMODEL + m;
    const size_t r1 = r0 + DMODEL;
#pragma unroll
    for (int dt = 0; dt < 4; dt++) {
      const int d0 = ((wid << 2) + dt) << 4;
      float xb0 = in0 ? Xn[r0 + d0] : 0.0f;
      float xb1 = in1 ? Xn[r1 + d0] : 0.0f;
      float kb0 = in0 ? Kn[r0 + d0] : 0.0f;
      float kb1 = in1 ? Kn[r1 + d0] : 0.0f;
      v2f bx = {xb0, xb1};
      v2f bk2 = {kb0, kb1};
      nacc[dt] = wmma4(a, bx,  nacc[dt]);
      dacc[dt] = wmma4(a, bk2, dacc[dt]);
    }
  }

  const size_t base = ((size_t)n * SL + I) * DMODEL;
#pragma unroll
  for (int dt = 0; dt < 4; dt++) {
    const int d = (((wid << 2) + dt) << 4) + m;
    const float sk = SK[n * DMODEL + d];
    const float sx = SX[n * DMODEL + d];
#pragma unroll
    for (int r = 0; r < 8; r++) {
      const int row = r + 8 * kh;
      const size_t idx = base + (size_t)row * DMODEL + d;
      float num = sx + nacc[dt][r];
      float den = sk + dacc[dt][r];
      y[idx] = sq[idx] * num / den;
    }
  }
}

// ---------------------------------------------------------------------------
// Kernel 5: output projection y @ Wo + bo, stored transposed [N,L,C]->[N,C,L]
// ---------------------------------------------------------------------------
__global__ __launch_bounds__(32) void k_out_proj(
    const float* __restrict__ y, const float* __restrict__ Wo,
    const float* __restrict__ bo, float* __restrict__ out) {
  const int lane = threadIdx.x & 31;
  const int m  = lane & 15;
  const int kh = lane >> 4;
  const int rt = blockIdx.x;
  const int n  = rt >> 7;
  const int l0 = (rt & (NTILE - 1)) << 4;

  const float* yn = y + ((size_t)n * SL + l0) * DMODEL;

  v8f acc[8] = {};
  for (int k0 = 0; k0 < DMODEL; k0 += 4) {
    // A fragment from y (row-major [L,128]): adjacent k -> one b64 load
    v2f a = *(const v2f*)(yn + (size_t)m * DMODEL + k0 + 2 * kh);
    const float* w0 = Wo + (size_t)(k0 + 2 * kh) * DMODEL + m;
    const float* w1 = w0 + DMODEL;
#pragma unroll
    for (int ct = 0; ct < 8; ct++) {
      v2f b = {w0[ct * 16], w1[ct * 16]};
      acc[ct] = wmma4(a, b, acc[ct]);
    }
  }

  float* ob = out + (size_t)n * DMODEL * SL;
#pragma unroll
  for (int ct = 0; ct < 8; ct++) {
    const int c = ct * 16 + m;
    const float bias = bo[c];
#pragma unroll
    for (int r = 0; r < 8; r++) {
      const int row = r + 8 * kh;
      ob[(size_t)c * SL + l0 + row] = acc[ct][r] + bias;   // transposed store
    }
  }
}

// ---------------------------------------------------------------------------
extern "C" void kernel_launch(void* const* d_in, const int* in_sizes, int n_in,
                              void* d_out, int out_size, void* d_ws, size_t ws_size,
                              hipStream_t stream) {
  (void)in_sizes; (void)n_in; (void)out_size; (void)ws_size;

  const float* x  = (const float*)d_in[0];
  const float* Wq = (const float*)d_in[1];
  const float* bq = (const float*)d_in[2];
  const float* Wk = (const float*)d_in[3];
  const float* bk = (const float*)d_in[4];
  const float* Wv = (const float*)d_in[5];
  const float* bv = (const float*)d_in[6];
  const float* Wo = (const float*)d_in[7];
  const float* bo = (const float*)d_in[8];
  const float* pb = (const float*)d_in[9];
  float* out = (float*)d_out;

  const size_t NLD = (size_t)NB * SL * DMODEL;
  char* w = (char*)d_ws;
  float* sq = (float*)w; w += NLD * sizeof(float);                 // sigmoid(q)
  float* ek = (float*)w; w += NLD * sizeof(float);                 // exp(k)
  float* Xv = (float*)w; w += NLD * sizeof(float);                 // exp(k)*v
  float* yb = (float*)w; w += NLD * sizeof(float);                 // pre-out-proj
  float* Wb = (float*)w; w += (size_t)NTILE * 16 * BAND_K * sizeof(float);
  float* PK = (float*)w; w += (size_t)NB * NTILE * DMODEL * sizeof(float);
  float* PX = (float*)w; w += (size_t)NB * NTILE * DMODEL * sizeof(float);
  float* SK = (float*)w; w += (size_t)NB * DMODEL * sizeof(float);
  float* SX = (float*)w; w += (size_t)NB * DMODEL * sizeof(float);

  const int nRowTiles = NB * NTILE;   // 4096

  k_band_w   <<<NTILE, 256, 0, stream>>>(pb, Wb);
  k_proj_qkv <<<nRowTiles, 96, 0, stream>>>(x, Wq, bq, Wk, bk, Wv, bv,
                                            sq, ek, Xv, PK, PX);
  k_colsum   <<<(NB * DMODEL + 255) / 256, 256, 0, stream>>>(PK, PX, SK, SX);
  k_band_attn<<<nRowTiles, 64, 0, stream>>>(Wb, Xv, ek, sq, SK, SX, yb);
  k_out_proj <<<nRowTiles, 32, 0, stream>>>(yb, Wo, bo, out);
}